// TriglialReservoir_20890720928327
// MI455X (gfx1250) — compile-verified
//
#include <hip/hip_runtime.h>
#include <hip/hip_bf16.h>
#include <stdint.h>

// ---------------------------------------------------------------------------
// Triglial reservoir, restructured for MI455X (gfx1250):
//  * input GEMM hoisted out of the scan -> V_WMMA_F32_16X16X4_F32
//  * microglia mask provably == 1 for 100 steps (0.99^99 > 0.01) -> dropped
//  * spike history as bitmask in LDS; delays packed to u8; W+delays (80MB)
//    are L2-resident (192MB) -> recurrent scan runs out of L2
// ---------------------------------------------------------------------------

#define STEPS   100
#define IN_DIM  512
#define HID     4096
#define OUT     64
#define MAX_DELAY 10
#define DECAY_F 0.951229424500714f   // exp(-1/20)
#define HIST_WORDS (HID / 32)        // 128 words per slot
#define ICHUNKS 32                   // i-dimension split for partial sums

typedef __attribute__((ext_vector_type(2))) float v2f;
typedef __attribute__((ext_vector_type(8))) float v8f;

// ---------------- init state ----------------
__global__ void init_state(float* v, float* gamma, float* rate, float* vsum,
                           unsigned* hist) {
    int j = blockIdx.x * 256 + threadIdx.x;     // 4096 threads
    v[j]     = 0.0f;
    gamma[j] = 1.0f;
    rate[j]  = 0.0f;
    vsum[j]  = 0.0f;
    if (j < HIST_WORDS * MAX_DELAY) hist[j] = 0u;
}

// ---------------- pack delays int32 -> u8 ----------------
__global__ void pack_delays(const int* __restrict__ delays,
                            unsigned char* __restrict__ d8) {
    size_t total = (size_t)HID * HID;
    size_t stride = (size_t)gridDim.x * blockDim.x;
    for (size_t idx = (size_t)blockIdx.x * blockDim.x + threadIdx.x;
         idx < total; idx += stride)
        d8[idx] = (unsigned char)delays[idx];
}

// ---------------- input GEMM: [STEPS,512] @ [512,4096] via f32 WMMA --------
// grid = (256 n-tiles, 7 m-tiles), block = 32 (one wave, EXEC all ones).
// Rows 100..111 of the padded output are never read, so out-of-range M rows
// are CLAMPED (not masked) -> no per-load EXEC juggling in the inner loop.
__global__ void input_gemm(const float* __restrict__ X,
                           const float* __restrict__ Win,
                           float* __restrict__ i_in) {
    int lane = threadIdx.x;          // 0..31
    int half = lane >> 4;            // 0 or 1
    int lmod = lane & 15;
    int nt = blockIdx.x;             // 0..255
    int mt = blockIdx.y;             // 0..6   (112 padded rows)
    int m = mt * 16 + lmod;
    int mc = (m < STEPS) ? m : (STEPS - 1);   // clamp: pollutes only unused rows
    int n = nt * 16 + lmod;

    const float* xrow = X + (size_t)mc * IN_DIM + 2 * half;
    const float* bcol = Win + n + (size_t)2 * half * HID;

    v8f c = {0.f, 0.f, 0.f, 0.f, 0.f, 0.f, 0.f, 0.f};
    for (int k0 = 0; k0 < IN_DIM; k0 += 4) {
        // A frag: lanes 0-15 hold {K0,K1}, lanes 16-31 hold {K2,K3}; 8B aligned
        v2f a = *(const v2f*)(xrow + k0);
        v2f b;
        b.x = bcol[(size_t)k0 * HID];
        b.y = bcol[(size_t)k0 * HID + HID];
        c = __builtin_amdgcn_wmma_f32_16x16x4_f32(
                /*neg_a=*/false, a, /*neg_b=*/false, b,
                /*c_mod=*/(short)0, c, /*reuse_a=*/false, /*reuse_b=*/false);
    }
    // D layout: VGPR r, lanes 0-15 -> M=r ; lanes 16-31 -> M=r+8
    for (int r = 0; r < 8; ++r) {
        int row = mt * 16 + r + 8 * half;    // < 112, padded buffer
        i_in[(size_t)row * HID + n] = c[r];
    }
}

// ---------------- recurrent current (per step) ----------------
// grid = (16 j-tiles, 32 i-chunks), block = 256.
// hist layout: [word 0..127][slot 0..9]  -> slot gather hits distinct banks.
// Fully branchless inner loop; tm = t % 10 precomputed on host.
__global__ void rec_current(const float* __restrict__ W,
                            const unsigned char* __restrict__ d8,
                            const unsigned* __restrict__ hist_g,
                            float* __restrict__ partial, int t, int tm) {
    __shared__ unsigned hist[HIST_WORDS * MAX_DELAY];  // 5 KB
    for (int idx = threadIdx.x; idx < HIST_WORDS * MAX_DELAY; idx += 256)
        hist[idx] = hist_g[idx];
    __syncthreads();

    int j  = blockIdx.x * 256 + threadIdx.x;
    int i0 = blockIdx.y * (HID / ICHUNKS);
    const float*         wp = W  + (size_t)i0 * HID + j;
    const unsigned char* dp = d8 + (size_t)i0 * HID + j;

    float acc0 = 0.0f, acc1 = 0.0f;
#pragma unroll 4
    for (int ii = 0; ii < HID / ICHUNKS; ++ii) {
        int i = i0 + ii;
        if ((ii & 3) == 0)   // static after unroll: prefetch W 16 rows ahead
            __builtin_prefetch(wp + (size_t)16 * HID, 0, 0);
        float w = wp[(size_t)ii * HID];
        int   d = dp[(size_t)ii * HID];         // 1..9
        int   slot = tm - d;
        slot += (slot < 0) ? 10 : 0;            // == (t - d) mod 10, always valid
        unsigned word = hist[(i >> 5) * MAX_DELAY + slot];
        bool on = (((word >> (i & 31)) & 1u) != 0u) & (d <= t);
        if (ii & 1) acc1 += on ? w : 0.0f;
        else        acc0 += on ? w : 0.0f;
    }
    partial[(size_t)blockIdx.y * HID + j] = acc0 + acc1;
}

// ---------------- LIF + astrocyte update (per step) ----------------
// grid = 16 blocks x 256; consecutive 32 threads form one wave -> ballot word.
__global__ void lif_step(const float* __restrict__ i_in_t,
                         const float* __restrict__ partial,
                         float* __restrict__ v, float* __restrict__ gamma,
                         float* __restrict__ rate, float* __restrict__ vsum,
                         unsigned* __restrict__ hist, int tm) {
    int j = blockIdx.x * 256 + threadIdx.x;

    float ir = 0.0f;
#pragma unroll
    for (int c = 0; c < ICHUNKS; ++c) ir += partial[(size_t)c * HID + j];

    float g0   = gamma[j];
    float itot = (i_in_t[j] + ir) * g0;
    float vn   = v[j] * DECAY_F + itot;
    bool  sp   = (vn >= 1.0f);
    float vout = sp ? 0.0f : vn;
    v[j]    = vout;
    vsum[j] += vout;                       // membrane trace accumulator

    float r = 0.9f * rate[j] + (sp ? 0.1f : 0.0f);
    rate[j] = r;
    float g = g0 + 0.01f * (0.1f - r);
    gamma[j] = fminf(2.0f, fmaxf(0.5f, g));

    unsigned m = __builtin_amdgcn_ballot_w32(sp);
    if ((threadIdx.x & 31) == 0)
        hist[(j >> 5) * MAX_DELAY + tm] = m;
}

// ---------------- readout: out = (vsum/STEPS) @ Wr + b ----------------
__global__ void readout(const float* __restrict__ vsum,
                        const float* __restrict__ Wr,
                        const float* __restrict__ b,
                        float* __restrict__ out) {
    __shared__ float red[256];
    int o = blockIdx.x;                    // 0..63
    float p = 0.0f;
    for (int j = threadIdx.x; j < HID; j += 256)
        p += vsum[j] * Wr[(size_t)j * OUT + o];
    red[threadIdx.x] = p;
    __syncthreads();
    for (int st = 128; st > 0; st >>= 1) {
        if (threadIdx.x < st) red[threadIdx.x] += red[threadIdx.x + st];
        __syncthreads();
    }
    if (threadIdx.x == 0) out[o] = b[o] + red[0] * (1.0f / (float)STEPS);
}

// ---------------------------------------------------------------------------
extern "C" void kernel_launch(void* const* d_in, const int* in_sizes, int n_in,
                              void* d_out, int out_size, void* d_ws, size_t ws_size,
                              hipStream_t stream) {
    const float* x    = (const float*)d_in[0];   // [100,512]
    const float* Win  = (const float*)d_in[1];   // [512,4096]
    const float* Wrec = (const float*)d_in[2];   // [4096,4096]
    const float* Wro  = (const float*)d_in[3];   // [4096,64]
    const float* bro  = (const float*)d_in[4];   // [64]
    const int*   dly  = (const int*)d_in[5];     // [4096,4096]

    // workspace carve-up (~18.5 MB total)
    char* ws = (char*)d_ws;
    unsigned char* d8 = (unsigned char*)ws;           ws += (size_t)HID * HID;            // 16 MB
    float* i_in    = (float*)ws;                      ws += (size_t)112 * HID * 4;        // 1.75 MB (112 padded rows)
    float* partial = (float*)ws;                      ws += (size_t)ICHUNKS * HID * 4;    // 512 KB
    unsigned* hist = (unsigned*)ws;                   ws += (size_t)HIST_WORDS * MAX_DELAY * 4 + 256;
    float* v       = (float*)ws;                      ws += (size_t)HID * 4;
    float* gamma   = (float*)ws;                      ws += (size_t)HID * 4;
    float* rate    = (float*)ws;                      ws += (size_t)HID * 4;
    float* vsum    = (float*)ws;                      ws += (size_t)HID * 4;

    init_state<<<HID / 256, 256, 0, stream>>>(v, gamma, rate, vsum, hist);
    pack_delays<<<4096, 256, 0, stream>>>(dly, d8);
    input_gemm<<<dim3(HID / 16, 112 / 16), 32, 0, stream>>>(x, Win, i_in);

    for (int t = 0; t < STEPS; ++t) {
        int tm = t % MAX_DELAY;
        rec_current<<<dim3(HID / 256, ICHUNKS), 256, 0, stream>>>(Wrec, d8, hist,
                                                                  partial, t, tm);
        lif_step<<<HID / 256, 256, 0, stream>>>(i_in + (size_t)t * HID, partial,
                                                v, gamma, rate, vsum, hist, tm);
    }

    readout<<<OUT, 256, 0, stream>>>(vsum, Wro, bro, (float*)d_out);
}